// ExpansionContrastModule_60181081751957
// MI455X (gfx1250) — compile-verified
//
#include <hip/hip_runtime.h>
#include <hip/hip_bf16.h>

// ---------------------------------------------------------------------------
// ExpansionContrastModule for MI455X (gfx1250, wave32, WMMA).
//
// Fusion: the 8-direction "diff conv" is center - shifted neighbor, so the
// giant [b,C,24,65536] "values" tensor is never materialized. All 1x1 convs
// (query/key/value/out) are M=32,K=32,N=pixels GEMMs done with
// V_WMMA_F32_16X16X4_F32 (f32 path; workload is HBM-bound at 23.3 TB/s so
// low-precision WMMA buys nothing and f32 preserves accuracy).
// ---------------------------------------------------------------------------

#define CCH   32          // channels
#define NSH   3           // shifts (1,2,4)
#define NDIR  8           // diff directions
#define NS    24          // 8*3 score entries
#define WSP   256         // full-res spatial dim
#define WSPD  64          // downsampled spatial dim (stride-4 conv)
#define NPIX  (WSP*WSP)   // 65536
#define NPIXD (WSPD*WSPD) // 4096

typedef float v2f __attribute__((ext_vector_type(2)));
typedef float v8f __attribute__((ext_vector_type(8)));

// diff-kernel direction offsets: d1 kernels then spatially-flipped d2 kernels
__device__ __constant__ int c_dy[NDIR] = {-1,-1,-1, 0,  1, 1, 1, 0};
__device__ __constant__ int c_dx[NDIR] = {-1, 0, 1, 1,  1, 0,-1,-1};

// ---------------------------------------------------------------------------
// 1) trans_layer: 5x5 conv, stride 4, pad 2, + bias.  [2,32,256,256]->[2,32,64,64]
// ---------------------------------------------------------------------------
__global__ void trans_conv_kernel(const float* __restrict__ cen,
                                  const float* __restrict__ W,   // [32,32,5,5]
                                  const float* __restrict__ bias,
                                  float* __restrict__ kv)        // [2,32,64,64]
{
    int idx = blockIdx.x * blockDim.x + threadIdx.x;   // 2*32*64*64 threads
    int x  = idx & (WSPD - 1);
    int y  = (idx >> 6) & (WSPD - 1);
    int co = (idx >> 12) & (CCH - 1);
    int b  = idx >> 17;

    const float* cb = cen + (size_t)b * CCH * NPIX;
    float acc = bias[co];
    for (int ci = 0; ci < CCH; ++ci) {
        const float* cp = cb + (size_t)ci * NPIX;
        const float* wp = W + ((size_t)(co * CCH + ci)) * 25;
#pragma unroll
        for (int ky = 0; ky < 5; ++ky) {
            int iy = 4 * y + ky - 2;
            if (iy < 0 || iy >= WSP) continue;
#pragma unroll
            for (int kx = 0; kx < 5; ++kx) {
                int ix = 4 * x + kx - 2;
                if (ix < 0 || ix >= WSP) continue;
                acc = fmaf(wp[ky * 5 + kx], cp[iy * WSP + ix], acc);
            }
        }
    }
    kv[idx] = acc;
}

// ---------------------------------------------------------------------------
// 2) Generic 1x1-conv GEMM via WMMA f32 16x16x4:
//      Y[z][b][m][n] = sum_k W[z][m][k] * X[b][k][n],  M=K=32.
//    Block = 256 threads = 8 waves; each wave owns one 16(m) x 16(n) D tile.
//    Optional fused BN(eval)+ReLU epilogue (for the final out_conv).
// ---------------------------------------------------------------------------
__global__ void gemm1x1_wmma(const float* __restrict__ W,   // [gridDim.z][32][32]
                             const float* __restrict__ X,   // [gridDim.y][32][npix]
                             float* __restrict__ Y,
                             int npix,
                             long long yBatchStride,        // elems
                             long long yZStride,            // elems
                             const float* __restrict__ bnG,
                             const float* __restrict__ bnB,
                             const float* __restrict__ bnM,
                             const float* __restrict__ bnV)
{
    const int lane = threadIdx.x & 31;
    const int wave = threadIdx.x >> 5;
    const int half = lane >> 4;          // 0: lanes 0-15, 1: lanes 16-31
    const int l    = lane & 15;
    const int m0   = (wave & 1) * 16;                    // m tile: 0 or 16
    const int n0   = blockIdx.x * 64 + (wave >> 1) * 16; // 4 n-tiles per block

    const float* Wm = W + (size_t)blockIdx.z * (CCH * CCH);
    const float* Xb = X + (size_t)blockIdx.y * CCH * (size_t)npix;
    float* Yb = Y + (size_t)blockIdx.y * (size_t)yBatchStride
                  + (size_t)blockIdx.z * (size_t)yZStride;

    v8f acc = {};
#pragma unroll
    for (int k0 = 0; k0 < CCH; k0 += 4) {
        const int ka = k0 + half * 2;    // this half-wave covers K = ka, ka+1
        v2f a, b;
        // A (16x4): lane m = l; VGPR0 -> K=ka, VGPR1 -> K=ka+1
        a.x = Wm[(m0 + l) * CCH + ka];
        a.y = Wm[(m0 + l) * CCH + ka + 1];
        // B (4x16): lane n = n0 + l, mirrored K split across half-waves
        b.x = Xb[(size_t)ka       * npix + n0 + l];
        b.y = Xb[(size_t)(ka + 1) * npix + n0 + l];
        acc = __builtin_amdgcn_wmma_f32_16x16x4_f32(
                  false, a, false, b, (short)0, acc, false, false);
    }

#pragma unroll
    for (int r = 0; r < 8; ++r) {
        int   m = m0 + half * 8 + r;     // D VGPR r -> M = r (+8 for hi half)
        float v = acc[r];
        if (bnG) {                        // fused BN (running stats) + ReLU
            float inv = rsqrtf(bnV[m] + 1e-5f);
            v = (v - bnM[m]) * (bnG[m] * inv) + bnB[m];
            v = fmaxf(v, 0.0f);
        }
        Yb[(size_t)m * npix + n0 + l] = v;
    }
}

// ---------------------------------------------------------------------------
// 3) score: per (b,c) block computes
//      qq = ||q||^2, and for the 24 (dir j, shift i) combos:
//      kk_s = ||k_i - shift(k_i)||^2 , qk_s = <q, k_i - shift(k_i)>
//    then score = l2norm_24( qk / (max(||q||,eps)*max(||k||,eps)) ).
//    Reductions via LDS float atomics (ds_add_f32).
// ---------------------------------------------------------------------------
__global__ void score_kernel(const float* __restrict__ q,     // [2][32][4096]
                             const float* __restrict__ kbuf,  // [3][2][32][4096]
                             float* __restrict__ score)       // [2][32][24]
{
    const int bc = blockIdx.x;                 // b*32 + c
    const size_t plane = (size_t)bc * NPIXD;
    const float* qp = q + plane;

    __shared__ float s_qq;
    __shared__ float s_kk[NS];
    __shared__ float s_qk[NS];
    __shared__ float s_sc[NS];
    if (threadIdx.x == 0) s_qq = 0.0f;
    if (threadIdx.x < NS) { s_kk[threadIdx.x] = 0.0f; s_qk[threadIdx.x] = 0.0f; }
    __syncthreads();

    float qq = 0.0f, kk[NS], qk[NS];
#pragma unroll
    for (int s = 0; s < NS; ++s) { kk[s] = 0.0f; qk[s] = 0.0f; }

    for (int p = threadIdx.x; p < NPIXD; p += blockDim.x) {
        int y = p >> 6, x = p & (WSPD - 1);
        float qv = qp[p];
        qq = fmaf(qv, qv, qq);
        for (int i = 0; i < NSH; ++i) {
            const float* kp = kbuf + (size_t)i * (2 * CCH * NPIXD) + plane;
            int d = 1 << i;                    // shifts 1,2,4
            float kc = kp[p];
#pragma unroll
            for (int j = 0; j < NDIR; ++j) {
                int yy = y + c_dy[j] * d, xx = x + c_dx[j] * d;
                float kn = (yy >= 0 && yy < WSPD && xx >= 0 && xx < WSPD)
                               ? kp[yy * WSPD + xx] : 0.0f;
                float ks = kc - kn;
                int s = j * NSH + i;
                kk[s] = fmaf(ks, ks, kk[s]);
                qk[s] = fmaf(qv, ks, qk[s]);
            }
        }
    }
    atomicAdd(&s_qq, qq);
#pragma unroll
    for (int s = 0; s < NS; ++s) { atomicAdd(&s_kk[s], kk[s]); atomicAdd(&s_qk[s], qk[s]); }
    __syncthreads();

    if (threadIdx.x < NS) {
        float qn = fmaxf(sqrtf(s_qq), 1e-12f);
        float kn = fmaxf(sqrtf(s_kk[threadIdx.x]), 1e-12f);
        s_sc[threadIdx.x] = s_qk[threadIdx.x] / (qn * kn);
    }
    __syncthreads();
    if (threadIdx.x < NS) {
        float ss = 0.0f;
#pragma unroll
        for (int s = 0; s < NS; ++s) ss = fmaf(s_sc[s], s_sc[s], ss);
        float nrm = fmaxf(sqrtf(ss), 1e-12f);
        score[(size_t)bc * NS + threadIdx.x] = s_sc[threadIdx.x] / nrm;
    }
}

// ---------------------------------------------------------------------------
// 4) combine: out_pre[b,c,p] = sum_i A_i * v_i(p) - sum_{i,j} sc[j*3+i] * v_i(p+off)
//    where A_i = sum_j sc[j*3+i].  (== score @ values without materializing it)
// ---------------------------------------------------------------------------
__global__ void combine_kernel(const float* __restrict__ vbuf,  // [3][2][32][65536]
                               const float* __restrict__ score, // [2][32][24]
                               float* __restrict__ outp)        // [2][32][65536]
{
    const int plane = blockIdx.y;              // b*32 + c
    __shared__ float sc[NS];
    __shared__ float sa[NSH];
    if (threadIdx.x < NS) sc[threadIdx.x] = score[(size_t)plane * NS + threadIdx.x];
    __syncthreads();
    if (threadIdx.x < NSH) {
        float a = 0.0f;
#pragma unroll
        for (int j = 0; j < NDIR; ++j) a += sc[j * NSH + threadIdx.x];
        sa[threadIdx.x] = a;
    }
    __syncthreads();

    int p = blockIdx.x * blockDim.x + threadIdx.x;
    int y = p >> 8, x = p & (WSP - 1);
    float acc = 0.0f;
    for (int i = 0; i < NSH; ++i) {
        const float* vp = vbuf + (size_t)i * (2 * CCH * NPIX) + (size_t)plane * NPIX;
        acc = fmaf(sa[i], vp[p], acc);
        int d = 1 << i;
#pragma unroll
        for (int j = 0; j < NDIR; ++j) {
            int yy = y + c_dy[j] * d, xx = x + c_dx[j] * d;
            if (yy >= 0 && yy < WSP && xx >= 0 && xx < WSP)
                acc = fmaf(-sc[j * NSH + i], vp[yy * WSP + xx], acc);
        }
    }
    outp[(size_t)plane * NPIX + p] = acc;
}

// ---------------------------------------------------------------------------
extern "C" void kernel_launch(void* const* d_in, const int* in_sizes, int n_in,
                              void* d_out, int out_size, void* d_ws, size_t ws_size,
                              hipStream_t stream)
{
    (void)in_sizes; (void)n_in; (void)out_size; (void)ws_size;

    const float* cen     = (const float*)d_in[0];  // [2,32,256,256]
    const float* trans_W = (const float*)d_in[1];  // [32,32,5,5]
    const float* trans_b = (const float*)d_in[2];  // [32]
    const float* query_W = (const float*)d_in[3];  // [32,32,1,1]
    const float* value_W = (const float*)d_in[4];  // [3,32,32,1,1]
    const float* key_W   = (const float*)d_in[5];  // [3,32,32,1,1]
    const float* out_W   = (const float*)d_in[6];  // [32,32,1,1]
    const float* bn_g    = (const float*)d_in[7];
    const float* bn_b    = (const float*)d_in[8];
    const float* bn_m    = (const float*)d_in[9];
    const float* bn_v    = (const float*)d_in[10];

    // workspace layout (f32 elements)
    float* kv    = (float*)d_ws;            // 2*32*4096      = 262144
    float* qbuf  = kv    + 2 * CCH * NPIXD; // 262144
    float* kbuf  = qbuf  + 2 * CCH * NPIXD; // 3*262144
    float* scbuf = kbuf  + NSH * 2 * CCH * NPIXD; // 1536 (pad to 2048)
    float* vbuf  = scbuf + 2048;            // 3*2*32*65536 = 12582912
    float* outp  = vbuf  + (size_t)NSH * 2 * CCH * NPIX; // 4194304
    // total ~72.4 MB

    // 1) trans conv (5x5, s4, p2) + bias
    trans_conv_kernel<<<dim3((2 * CCH * NPIXD) / 256), 256, 0, stream>>>(
        cen, trans_W, trans_b, kv);

    // 2) query = query_W @ kv   [2,32,4096]
    gemm1x1_wmma<<<dim3(NPIXD / 64, 2, 1), 256, 0, stream>>>(
        query_W, kv, qbuf, NPIXD, (long long)CCH * NPIXD, 0,
        nullptr, nullptr, nullptr, nullptr);

    // 3) keys_i = key_W[i] @ kv  [3][2,32,4096]
    gemm1x1_wmma<<<dim3(NPIXD / 64, 2, NSH), 256, 0, stream>>>(
        key_W, kv, kbuf, NPIXD, (long long)CCH * NPIXD,
        (long long)2 * CCH * NPIXD, nullptr, nullptr, nullptr, nullptr);

    // 4) values_i = value_W[i] @ cen  [3][2,32,65536]
    gemm1x1_wmma<<<dim3(NPIX / 64, 2, NSH), 256, 0, stream>>>(
        value_W, cen, vbuf, NPIX, (long long)CCH * NPIX,
        (long long)2 * CCH * NPIX, nullptr, nullptr, nullptr, nullptr);

    // 5) cosine scores + l2norm over the 24 entries
    score_kernel<<<dim3(2 * CCH), 256, 0, stream>>>(qbuf, kbuf, scbuf);

    // 6) out_pre = score @ values (fused diffs, no materialized values tensor)
    combine_kernel<<<dim3(NPIX / 256, 2 * CCH), 256, 0, stream>>>(vbuf, scbuf, outp);

    // 7) out = ReLU(BN(out_W @ out_pre))  -> d_out
    gemm1x1_wmma<<<dim3(NPIX / 64, 2, 1), 256, 0, stream>>>(
        out_W, outp, (float*)d_out, NPIX, (long long)CCH * NPIX, 0,
        bn_g, bn_b, bn_m, bn_v);
}